// Blob_DC_and_BCE_loss_19275813225170
// MI455X (gfx1250) — compile-verified
//
#include <hip/hip_runtime.h>

#define DD     128
#define NV     (DD*DD*DD)        /* 2,097,152 voxels per sample (2^21) */
#define BB     2
#define NT     (BB*NV)
#define KMAX   8
#define SMOOTHF 1e-5f
#define LN2F   0.69314718055994531f

typedef float v2f __attribute__((ext_vector_type(2)));
typedef float v8f __attribute__((ext_vector_type(8)));
typedef unsigned int v4u __attribute__((ext_vector_type(4)));
typedef int   v4i __attribute__((ext_vector_type(4)));
typedef int   v8i __attribute__((ext_vector_type(8)));

// ---------------------------------------------------------------------------
// Label init: tgt label = local_idx+1 where target>0.5, out label where x>=0.
// ---------------------------------------------------------------------------
__global__ void init_labels(const float* __restrict__ x, const float* __restrict__ y,
                            int* __restrict__ tgt, int* __restrict__ outc) {
  int v = blockIdx.x * 256 + threadIdx.x;
  int loc = v & (NV - 1);
  tgt[v]  = (y[v] > 0.5f)  ? (loc + 1) : 0;
  outc[v] = (x[v] >= 0.0f) ? (loc + 1) : 0;
}

// ---------------------------------------------------------------------------
// One global CC relaxation pass: 8x8x8 tile + halo in LDS, 8 local iterations.
// Interior tiles fetch the 10x10x10 halo'd tile with the Tensor Data Mover.
// ---------------------------------------------------------------------------
__device__ __forceinline__ int relax6(const int* lab, int i) {
  int c = lab[i];
  if (c == 0) return 0;
  c = max(c, lab[i - 1]);   c = max(c, lab[i + 1]);
  c = max(c, lab[i - 10]);  c = max(c, lab[i + 10]);
  c = max(c, lab[i - 100]); c = max(c, lab[i + 100]);
  return c;
}

__global__ void cc_prop(const int* __restrict__ src, int* __restrict__ dst) {
  __shared__ int lab[1000];                    // 10*10*10 halo'd tile
  const int tid = threadIdx.x;
  const int b   = blockIdx.x;
  const int s   = blockIdx.y;
  const int tx = b & 15, ty = (b >> 4) & 15, tz = b >> 8;
  const int gx0 = tx * 8 - 1, gy0 = ty * 8 - 1, gz0 = tz * 8 - 1;
  const int* sbase = src + (size_t)s * NV;
  const bool interior = (tx > 0) && (tx < 15) && (ty > 0) && (ty < 15) && (tz > 0) && (tz < 15);

  if (interior) {
    if (tid < 32) {                            // wave 0 drives the TDM
      unsigned long long ga =
          (unsigned long long)(const char*)(sbase + ((gz0 * DD + gy0) * DD + gx0));
      unsigned int ldsoff = (unsigned int)(unsigned long long)(void*)&lab[0];
      // D# group 0: count=1 | lds_addr | global_addr(57b) | type=2
      v4u g0;
      g0[0] = 1u;
      g0[1] = ldsoff;
      g0[2] = (unsigned int)(ga & 0xffffffffull);
      g0[3] = (unsigned int)((ga >> 32) & 0x1ffffffull) | (2u << 30);
      // D# group 1: data_size=4B, tensor 128x128(x128), tile 10x10x10,
      //             dim0_stride=128, dim1_stride=16384
      v8i g1;
      g1[0] = (2 << 16);                       // data_size = 4 bytes
      g1[1] = (DD << 16);                      // tensor_dim0[15:0]  @ bits 63:48
      g1[2] = (DD << 16);                      // tensor_dim1[15:0]  @ bits 95:80
      g1[3] = (10 << 16);                      // tile_dim0 = 10     @ bits 127:112
      g1[4] = 10 | (10 << 16);                 // tile_dim1, tile_dim2
      g1[5] = DD;                              // tensor_dim0_stride low32
      g1[6] = ((DD * DD) & 0xffff) << 16;      // tensor_dim1_stride[15:0]
      g1[7] = (DD * DD) >> 16;                 // tensor_dim1_stride[47:16]
      v4i g2; g2[0] = DD; g2[1] = 0; g2[2] = NV; g2[3] = 0;  // dim2=128, dim2_stride=NV
      v4i g3; g3[0] = 0;  g3[1] = 0; g3[2] = 0;  g3[3] = 0;
      // clang-23 lane: 6-arg form (uint32x4, int32x8, int32x4, int32x4, int32x8, cpol)
      v8i g23;
      g23[0] = g2[0]; g23[1] = g2[1]; g23[2] = g2[2]; g23[3] = g2[3];
      g23[4] = g3[0]; g23[5] = g3[1]; g23[6] = g3[2]; g23[7] = g3[3];
      __builtin_amdgcn_tensor_load_to_lds(g0, g1, g2, g3, g23, 0);
      __builtin_amdgcn_s_wait_tensorcnt(0);
    }
  } else {
    for (int i = tid; i < 1000; i += 256) {
      int lz = i / 100; int r = i - lz * 100; int ly = r / 10; int lx = r - ly * 10;
      int gz = gz0 + lz, gy = gy0 + ly, gx = gx0 + lx;
      int val = 0;
      if ((unsigned)gz < (unsigned)DD && (unsigned)gy < (unsigned)DD && (unsigned)gx < (unsigned)DD)
        val = sbase[(gz * DD + gy) * DD + gx];
      lab[i] = val;
    }
  }
  __syncthreads();

  const int e0 = tid, e1 = tid + 256;          // each thread owns 2 tile voxels
  const int z0 = e0 >> 6, y0 = (e0 >> 3) & 7, x0 = e0 & 7;
  const int z1 = e1 >> 6, y1 = (e1 >> 3) & 7, x1 = e1 & 7;
  const int i0 = (z0 + 1) * 100 + (y0 + 1) * 10 + (x0 + 1);
  const int i1 = (z1 + 1) * 100 + (y1 + 1) * 10 + (x1 + 1);

  for (int it = 0; it < 8; ++it) {
    int n0 = relax6(lab, i0);
    int n1 = relax6(lab, i1);
    __syncthreads();
    lab[i0] = n0; lab[i1] = n1;
    __syncthreads();
  }

  int* dbase = dst + (size_t)s * NV;
  dbase[((tz * 8 + z0) * DD + (ty * 8 + y0)) * DD + (tx * 8 + x0)] = lab[i0];
  dbase[((tz * 8 + z1) * DD + (ty * 8 + y1)) * DD + (tx * 8 + x1)] = lab[i1];
}

// ---------------------------------------------------------------------------
// segment_max: mapped[out_label] = max tgt label overlapping that blob.
// ---------------------------------------------------------------------------
__global__ void map_overlap(const int* __restrict__ tgt, const int* __restrict__ outc,
                            int* __restrict__ mapped) {
  int v = blockIdx.x * 256 + threadIdx.x;
  int s = v >> 21;
  int oc = outc[v];
  if (oc > 0) atomicMax(&mapped[(size_t)s * (NV + 1) + oc], tgt[v]);
}

// Representative voxel of each component holds its own index+1 as final label.
__global__ void find_labels(const int* __restrict__ tgt, int* __restrict__ cnt,
                            int* __restrict__ lblraw) {
  int v = blockIdx.x * 256 + threadIdx.x;
  int s = v >> 21;
  int t = tgt[v];
  if (t > 0 && t == (v & (NV - 1)) + 1) {
    int pos = atomicAdd(&cnt[s], 1);
    if (pos < KMAX) lblraw[s * KMAX + pos] = t;
  }
}

// ---------------------------------------------------------------------------
// Dense base sums (bce, p*y, p, y) per sample. Wave reduction via
// V_WMMA_F32_16X16X4_F32 with a parity-selector B (exact f32 math):
// D[m][n] = q(n%2)_m + q(n%2)_{m+16}, finished with 8 adds + 2 shuffles.
// ---------------------------------------------------------------------------
__global__ void base_sums(const float* __restrict__ x, const float* __restrict__ y,
                          float* __restrict__ bpart) {
  int tid = threadIdx.x, blk = blockIdx.x;
  int base = blk * 8192 + tid;
  float ab = 0.f, ai = 0.f, ap = 0.f, ag = 0.f;
  for (int j = 0; j < 32; ++j) {
    int v = base + j * 256;
    __builtin_prefetch(&x[v + 2048], 0, 0);
    __builtin_prefetch(&y[v + 2048], 0, 0);
    float xv = x[v], yv = y[v];
    float bce = fmaxf(xv, 0.f) - xv * yv + log1pf(__expf(-fabsf(xv)));
    float pv  = 1.f / (1.f + __expf(-xv));
    ab += bce; ai += pv * yv; ap += pv; ag += yv;
  }
  int lane = tid & 31, wid = tid >> 5;
  v2f a1; a1.x = ab; a1.y = ai;
  v2f a2; a2.x = ap; a2.y = ag;
  v2f bs;                                        // B col parity selector
  if (lane & 1) { bs.x = 0.f; bs.y = 1.f; } else { bs.x = 1.f; bs.y = 0.f; }
  v8f c0 = {0.f, 0.f, 0.f, 0.f, 0.f, 0.f, 0.f, 0.f};
  v8f d1 = __builtin_amdgcn_wmma_f32_16x16x4_f32(false, a1, false, bs, (short)0, c0, false, false);
  v8f d2 = __builtin_amdgcn_wmma_f32_16x16x4_f32(false, a2, false, bs, (short)0, c0, false, false);
  float t1 = d1[0] + d1[1] + d1[2] + d1[3] + d1[4] + d1[5] + d1[6] + d1[7];
  float t2 = d2[0] + d2[1] + d2[2] + d2[3] + d2[4] + d2[5] + d2[6] + d2[7];
  float sb = __shfl(t1, 0) + __shfl(t1, 16);
  float si = __shfl(t1, 1) + __shfl(t1, 17);
  float sp = __shfl(t2, 0) + __shfl(t2, 16);
  float sg = __shfl(t2, 1) + __shfl(t2, 17);
  __shared__ float wsum[8][4];
  if (lane == 0) { wsum[wid][0] = sb; wsum[wid][1] = si; wsum[wid][2] = sp; wsum[wid][3] = sg; }
  __syncthreads();
  if (tid < 4) {
    float acc = 0.f;
    for (int w = 0; w < 8; ++w) acc += wsum[w][tid];   // fixed order: deterministic
    bpart[blk * 4 + tid] = acc;
  }
}

// ---------------------------------------------------------------------------
// Sparse kill sums per label k: sum over killed voxels of
// (bce-ln2, p*y, p-0.5, y). Deterministic LDS tree reductions.
// ---------------------------------------------------------------------------
__global__ void kill_sums(const float* __restrict__ x, const float* __restrict__ y,
                          const int* __restrict__ tgt, const int* __restrict__ outc,
                          const int* __restrict__ mapped, const int* __restrict__ lbl,
                          float* __restrict__ kpart) {
  __shared__ int Ls[KMAX];
  __shared__ float red[256];
  int tid = threadIdx.x, blk = blockIdx.x;
  int s = blk >> 8;
  if (tid < KMAX) Ls[tid] = lbl[s * KMAX + tid];
  __syncthreads();
  float acc[KMAX * 4];
#pragma unroll
  for (int i = 0; i < KMAX * 4; ++i) acc[i] = 0.f;
  int base = blk * 8192 + tid;
  const int* map_s = mapped + (size_t)s * (NV + 1);
  for (int j = 0; j < 32; ++j) {
    int v = base + j * 256;
    int t = tgt[v];
    int oc = outc[v];
    int m = (oc > 0) ? map_s[oc] : 0;
    if ((t | m) != 0) {
      float xv = x[v], yv = y[v];
      float bce = fmaxf(xv, 0.f) - xv * yv + log1pf(__expf(-fabsf(xv)));
      float pv  = 1.f / (1.f + __expf(-xv));
#pragma unroll
      for (int k = 0; k < KMAX; ++k) {
        int L = Ls[k];
        bool killed = ((t != 0) && (t != L)) || ((m != 0) && (m != L));
        if (killed) {
          acc[k * 4 + 0] += bce - LN2F;
          acc[k * 4 + 1] += pv * yv;
          acc[k * 4 + 2] += pv - 0.5f;
          acc[k * 4 + 3] += yv;
        }
      }
    }
  }
#pragma unroll 1
  for (int kq = 0; kq < KMAX * 4; ++kq) {
    red[tid] = acc[kq];
    __syncthreads();
    for (int off = 128; off > 0; off >>= 1) {
      if (tid < off) red[tid] += red[tid + off];
      __syncthreads();
    }
    if (tid == 0) kpart[blk * 32 + kq] = red[0];
    __syncthreads();
  }
}

// ---------------------------------------------------------------------------
// Final combine: reduce block partials in fixed order, sort labels, compute
// 0.3*global + 0.7*blob loss exactly as the reference does.
// ---------------------------------------------------------------------------
__global__ void finalize_loss(const float* __restrict__ bpart, const float* __restrict__ kpart,
                              const int* __restrict__ cnt, const int* __restrict__ lblraw,
                              float* __restrict__ outp) {
  __shared__ float BS[2][4];
  __shared__ float KS[2][KMAX][4];
  __shared__ int   LB[2][KMAX];
  __shared__ int   NC[2];
  int t = threadIdx.x;
  if (t < 8) {
    int s = t >> 2, q = t & 3;
    float a = 0.f;
    for (int b = 0; b < 256; ++b) a += bpart[(s * 256 + b) * 4 + q];
    BS[s][q] = a;
  }
  if (t < 64) {
    int s = t >> 5, kq = t & 31;
    float a = 0.f;
    for (int b = 0; b < 256; ++b) a += kpart[(s * 256 + b) * 32 + kq];
    KS[s][kq >> 2][kq & 3] = a;
  }
  if (t < 2) {
    int tmp[KMAX];
    for (int i = 0; i < KMAX; ++i) tmp[i] = lblraw[t * KMAX + i];
    for (int i = 0; i < KMAX; ++i) {                      // selection sort desc
      int best = i;
      for (int j = i + 1; j < KMAX; ++j) if (tmp[j] > tmp[best]) best = j;
      int sw = tmp[i]; tmp[i] = tmp[best]; tmp[best] = sw;
    }
    int n = 0;
    for (int i = 0; i < KMAX; ++i) { LB[t][i] = tmp[i]; if (tmp[i] > 0) ++n; }
    NC[t] = n;
  }
  __syncthreads();
  if (t == 0) {
    const float invN = 1.f / (float)NV;
    float bceg = (BS[0][0] + BS[1][0]) * (0.5f * invN);
    float dcg  = (2.f * (BS[0][1] + BS[1][1]) + SMOOTHF) /
                 fmaxf(BS[0][2] + BS[1][2] + BS[0][3] + BS[1][3] + SMOOTHF, 1e-8f);
    float gl = bceg - dcg;

    float contrib = 0.f, count = 0.f;
    for (int s = 0; s < 2; ++s) {
      int n = NC[s];
      if (n > 1) {
        float c = 0.f;
        for (int k = 0; k < KMAX; ++k) {
          if (LB[s][k] > 0) {
            float bk = BS[s][0] - KS[s][k][0];
            float Ik = BS[s][1] - KS[s][k][1];
            float Pk = BS[s][2] - KS[s][k][2];
            float Gk = BS[s][3] - KS[s][k][3];
            c += bk * invN - (2.f * Ik + SMOOTHF) / fmaxf(Pk + Gk + SMOOTHF, 1e-8f);
          }
        }
        contrib += c; count += (float)n;
      } else {
        contrib += BS[s][0] * invN -
                   (2.f * BS[s][1] + SMOOTHF) / fmaxf(BS[s][2] + BS[s][3] + SMOOTHF, 1e-8f);
        count += 1.f;
      }
    }
    float blob = contrib / fmaxf(count, 1.f);
    outp[0] = 0.3f * gl + 0.7f * blob;
  }
}

// ---------------------------------------------------------------------------
extern "C" void kernel_launch(void* const* d_in, const int* in_sizes, int n_in,
                              void* d_out, int out_size, void* d_ws, size_t ws_size,
                              hipStream_t stream) {
  (void)in_sizes; (void)n_in; (void)out_size; (void)ws_size;
  const float* x = (const float*)d_in[0];   // net_output [2,1,128,128,128] f32
  const float* y = (const float*)d_in[1];   // target     [2,1,128,128,128] f32

  int* ws     = (int*)d_ws;
  int* tgt    = ws;                                   // NT ints  (tgt labels, ping)
  int* outc   = ws + (size_t)NT;                      // NT ints  (out labels, ping)
  int* pong   = ws + (size_t)2 * NT;                  // NT ints  (shared pong)
  int* mapped = ws + (size_t)3 * NT;                  // BB*(NV+1) ints
  int* smallb = ws + (size_t)3 * NT + (size_t)BB * (NV + 1);
  int* cnt    = smallb;                               // 2 ints
  int* lblraw = smallb + 2;                           // 16 ints
  float* bpart = (float*)(smallb + 2 + 16);           // 512*4 floats
  float* kpart = bpart + 512 * 4;                     // 512*32 floats
  float* outp  = (float*)d_out;

  // Zero only what must start at zero (mapped, cnt, labels); graph-capture safe.
  (void)hipMemsetAsync((void*)mapped, 0, ((size_t)BB * (NV + 1) + 18) * sizeof(int), stream);

  init_labels<<<NT / 256, 256, 0, stream>>>(x, y, tgt, outc);

  for (int it = 0; it < 48; ++it) {                   // target CC (ends in tgt)
    if ((it & 1) == 0) cc_prop<<<dim3(4096, BB), 256, 0, stream>>>(tgt, pong);
    else               cc_prop<<<dim3(4096, BB), 256, 0, stream>>>(pong, tgt);
  }
  for (int it = 0; it < 48; ++it) {                   // predicted CC (ends in outc)
    if ((it & 1) == 0) cc_prop<<<dim3(4096, BB), 256, 0, stream>>>(outc, pong);
    else               cc_prop<<<dim3(4096, BB), 256, 0, stream>>>(pong, outc);
  }

  map_overlap<<<NT / 256, 256, 0, stream>>>(tgt, outc, mapped);
  find_labels<<<NT / 256, 256, 0, stream>>>(tgt, cnt, lblraw);
  base_sums<<<512, 256, 0, stream>>>(x, y, bpart);
  kill_sums<<<512, 256, 0, stream>>>(x, y, tgt, outc, mapped, lblraw, kpart);
  finalize_loss<<<1, 64, 0, stream>>>(bpart, kpart, cnt, lblraw, outp);
}